// KGFlex_TFModel_712964571899
// MI455X (gfx1250) — compile-verified
//
#include <hip/hip_runtime.h>

// CDNA5 / gfx1250: wave32, WMMA f32 16x16x4.
typedef __attribute__((ext_vector_type(2))) float v2f;
typedef __attribute__((ext_vector_type(8))) float v8f;

#define FACTORS 64
#define NBLOCKS 512
#define NTHREADS 256  // 8 waves of 32

// Each wave processes 16 gathered (b,f) pairs at a time:
//   A = 16 rows H[u_n]   (16 x 64)
//   B = 16 rows G[f_n] laid out as columns (64 x 16)
//   C = A x B via 16 chained v_wmma_f32_16x16x4_f32; diag(C)[n] = H[u_n].G[f_n]
// contribution = sum_n K[u_n, f_n] * diag(C)[n]
__global__ void kgflex_partial_kernel(const int* __restrict__ user,
                                      const int* __restrict__ sample_idx,
                                      const int* __restrict__ feat_idx,
                                      const float* __restrict__ H,
                                      const float* __restrict__ G,
                                      const float* __restrict__ K,
                                      float* __restrict__ partial,
                                      int total, int kfeat) {
    const int lane      = threadIdx.x & 31;
    const int waveInBlk = threadIdx.x >> 5;
    const int wavesBlk  = blockDim.x >> 5;
    const int waveGlob  = blockIdx.x * wavesBlk + waveInBlk;
    const int numWaves  = gridDim.x * wavesBlk;
    const int numGroups = (total + 15) >> 4;

    const int n    = lane & 15;   // which of the 16 pairs this lane helps stage
    const int half = lane >> 4;   // K-halves: lanes 16-31 carry K offset +2

    float sum = 0.0f;

    // Wave-uniform grid-stride loop -> EXEC stays all-ones through WMMA.
    for (int g = waveGlob; g < numGroups; g += numWaves) {
        int idx   = g * 16 + n;
        bool valid = idx < total;
        int ic    = valid ? idx : (total - 1);   // clamp tail, weight zeroed below

        int b = sample_idx[ic];
        int f = feat_idx[ic];
        int u = user[b];

        const float* __restrict__ hrow = H + (size_t)u * FACTORS;
        const float* __restrict__ grow = G + (size_t)f * FACTORS;

        v8f acc = {0.f, 0.f, 0.f, 0.f, 0.f, 0.f, 0.f, 0.f};

        // A 16x4 f32 layout: lane L -> M=L&15, VGPR0=K0+2*(L>=16), VGPR1=+1.
        // B 4x16 f32 layout is symmetric: lane L -> N=L&15, same K offsets.
        // => both operands: float2 at row[4*s + 2*half].
        #pragma unroll
        for (int s = 0; s < FACTORS / 4; ++s) {
            int koff = 4 * s + 2 * half;
            v2f av; av.x = hrow[koff]; av.y = hrow[koff + 1];
            v2f bv; bv.x = grow[koff]; bv.y = grow[koff + 1];
            acc = __builtin_amdgcn_wmma_f32_16x16x4_f32(
                /*neg_a=*/false, av, /*neg_b=*/false, bv,
                /*c_mod=*/(short)0, acc, /*reuse_a=*/false, /*reuse_b=*/false);
        }

        // Gathered affinity weight (zero for tail-clamped lanes).
        float w = valid ? K[(size_t)u * (size_t)kfeat + (size_t)f] : 0.0f;

        // Diagonal of C: lanes 0-7 hold (M=lane,N=lane) in acc[lane];
        // lanes 24-31 hold (M=lane-16,N=lane-16) in acc[lane-24].
        int sel = lane & 7;
        float d = acc[0];
        d = (sel == 1) ? acc[1] : d;
        d = (sel == 2) ? acc[2] : d;
        d = (sel == 3) ? acc[3] : d;
        d = (sel == 4) ? acc[4] : d;
        d = (sel == 5) ? acc[5] : d;
        d = (sel == 6) ? acc[6] : d;
        d = (sel == 7) ? acc[7] : d;
        bool holdsDiag = (lane < 8) || (lane >= 24);
        sum += holdsDiag ? (d * w) : 0.0f;
    }

    // Wave32 reduction.
    #pragma unroll
    for (int off = 16; off > 0; off >>= 1)
        sum += __shfl_xor(sum, off, 32);

    __shared__ float ssum[NTHREADS / 32];
    if (lane == 0) ssum[waveInBlk] = sum;
    __syncthreads();
    if (threadIdx.x == 0) {
        float s = 0.0f;
        for (int i = 0; i < wavesBlk; ++i) s += ssum[i];
        partial[blockIdx.x] = s;
    }
}

__global__ void kgflex_final_reduce(const float* __restrict__ partial, int n,
                                    float* __restrict__ out) {
    __shared__ float ss[256];
    float s = 0.0f;
    for (int i = threadIdx.x; i < n; i += blockDim.x) s += partial[i];
    ss[threadIdx.x] = s;
    __syncthreads();
    for (int stride = 128; stride > 0; stride >>= 1) {
        if ((int)threadIdx.x < stride) ss[threadIdx.x] += ss[threadIdx.x + stride];
        __syncthreads();
    }
    if (threadIdx.x == 0) out[0] = ss[0];
}

extern "C" void kernel_launch(void* const* d_in, const int* in_sizes, int n_in,
                              void* d_out, int out_size, void* d_ws, size_t ws_size,
                              hipStream_t stream) {
    const int*   user       = (const int*)d_in[0];
    const int*   sample_idx = (const int*)d_in[1];
    const int*   feat_idx   = (const int*)d_in[2];
    const float* H          = (const float*)d_in[3];
    const float* G          = (const float*)d_in[4];
    const float* K          = (const float*)d_in[5];
    float* out = (float*)d_out;

    int total = in_sizes[1];                 // TOTAL_FEATS
    int kfeat = in_sizes[4] / FACTORS;       // NUM_FEATURES from G's flat size

    float* partial = (float*)d_ws;           // NBLOCKS floats (2 KB)

    kgflex_partial_kernel<<<NBLOCKS, NTHREADS, 0, stream>>>(
        user, sample_idx, feat_idx, H, G, K, partial, total, kfeat);
    kgflex_final_reduce<<<1, 256, 0, stream>>>(partial, NBLOCKS, out);
}